// MultiHeadGraphAttentionLayer_16372415332457
// MI455X (gfx1250) — compile-verified
//
#include <hip/hip_runtime.h>
#include <stdint.h>

// MI455X / gfx1250 multi-head graph-attention layer.
//   Stage 1: hp[b,h,n,d] = sum_f h[b,n,f] * W[h,f,d]             (fp32 WMMA 16x16x4)
//   Stage 2: flash-style online LeakyReLU-softmax row-sums of hp@hp^T.
//     - 128-row j-blocks staged to LDS via global_load_async_to_lds_b128
//       (ASYNCcnt path, double-buffered 2x32 KB), 4 waves/WG share each block.
//     - 8 independent WMMA accumulator chains per wave -> 128 WMMAs between syncs.
//     - Per-LANE online (max, sum-exp) stats; single cross-lane combine at the
//       end of the i-tile (no shuffles in the hot loop).
//   out = hp * rowsum(att)  (row-sum of softmax, ~1.0, computed faithfully).
// hp (16.8 MB) lives in d_ws -> L2-resident (192 MB). Score stage is the
// compute-bound part (~17.2 GFLOP fp32); HBM traffic ~50 MB ~= 2 us @ 23.3 TB/s.

typedef float v2f __attribute__((ext_vector_type(2)));
typedef float v8f __attribute__((ext_vector_type(8)));

#define B_    8
#define N_    2048
#define F_    256
#define H_    4
#define D_    64
#define HD_   (H_ * D_)
#define ALPHA 0.2f
#define JBLK  128             // j-rows staged per LDS block (32 KB)
#define NJB   (N_ / JBLK)
#define JT_   (JBLK / 16)     // 8 j-subtiles per staged block

// ---------------------------------------------------------------------------
// Stage 1: per-head projection. One wave -> 16(N-rows) x 64(D) tile.
// A 16x4 f32: lanes 0-15 row=lane K={k,k+1}; lanes 16-31 row=lane-16 K={k+2,k+3}.
// B 4x16 f32: col=lane&15; lanes 0-15 K-rows {k,k+1}; lanes 16-31 {k+2,k+3}.
// C/D 16x16: col=lane&15, row = vgpr + 8*(lane>>4).
// ---------------------------------------------------------------------------
__global__ __launch_bounds__(32) void gat_proj_wmma(const float* __restrict__ h,
                                                    const float* __restrict__ W,
                                                    float* __restrict__ hp)
{
    const int lane = threadIdx.x;
    const int half = lane >> 4;
    const int l16  = lane & 15;
    const int n0   = blockIdx.x * 16;
    const int hh   = blockIdx.y;
    const int b    = blockIdx.z;

    const float* hB = h + ((size_t)b * N_ + n0) * F_;
    const float* Wh = W + (size_t)hh * F_ * D_;

    v8f acc[4];
    for (int dt = 0; dt < 4; ++dt) acc[dt] = (v8f){0.f,0.f,0.f,0.f,0.f,0.f,0.f,0.f};

    for (int k = 0; k < F_; k += 4) {
        v2f a = *(const v2f*)(hB + (size_t)l16 * F_ + k + half * 2);
        #pragma unroll
        for (int dt = 0; dt < 4; ++dt) {
            v2f bm;
            bm.x = Wh[(size_t)(k + half * 2    ) * D_ + dt * 16 + l16];
            bm.y = Wh[(size_t)(k + half * 2 + 1) * D_ + dt * 16 + l16];
            acc[dt] = __builtin_amdgcn_wmma_f32_16x16x4_f32(
                false, a, false, bm, (short)0, acc[dt], false, false);
        }
    }

    float* hpT = hp + (((size_t)b * H_ + hh) * N_ + n0) * D_;
    #pragma unroll
    for (int dt = 0; dt < 4; ++dt)
        #pragma unroll
        for (int r = 0; r < 8; ++r)
            hpT[(size_t)(r + half * 8) * D_ + dt * 16 + l16] = acc[dt][r];
}

// ---------------------------------------------------------------------------
// Stage 2: 4 waves per block; wave w owns i-tile (blockIdx.x*4 + w).
// ---------------------------------------------------------------------------
__global__ __launch_bounds__(128) void gat_attn_rowsum_wmma(const float* __restrict__ hp,
                                                            float* __restrict__ out)
{
    __shared__ float jbuf[2][JBLK * D_];   // 2 x 32 KB double buffer
    __shared__ float fsh[4][16];

    const int tid  = threadIdx.x;
    const int wave = tid >> 5;
    const int lane = tid & 31;
    const int half = lane >> 4;
    const int l16  = lane & 15;
    const int hh   = blockIdx.y;
    const int b    = blockIdx.z;
    const int i0   = (blockIdx.x * 4 + wave) * 16;

    const float* hpBH = hp + (((size_t)b * H_ + hh) * N_) * D_;

    // Resident A fragments for this wave's i-tile (16 K-steps of 4 -> 32 VGPRs)
    v2f afrag[16];
    #pragma unroll
    for (int kk = 0; kk < 16; ++kk)
        afrag[kk] = *(const v2f*)(hpBH + (size_t)(i0 + l16) * D_ + kk * 4 + half * 2);

    // Per-LANE flash statistics: this lane's j-column slice, 8 rows per half-wave.
    float m[8], s[8];
    #pragma unroll
    for (int r = 0; r < 8; ++r) { m[r] = -__builtin_inff(); s[r] = 0.0f; }

    // Async copy of j-block jb (contiguous 32 KB of hp) into jbuf[bufsel].
    // 128 threads x 16 B x 16 sweeps; tracked by ASYNCcnt, overlaps WMMA issue.
    auto issue_jblock = [&](int bufsel, int jb) {
        const char* src = (const char*)(hpBH + (size_t)jb * JBLK * D_);
        unsigned ldsbase = (unsigned)(uintptr_t)(&jbuf[bufsel][0]);
        #pragma unroll
        for (int sweep = 0; sweep < 16; ++sweep) {
            const unsigned off = (unsigned)(sweep * 128 + tid) * 16u;
            unsigned lds = ldsbase + off;
            uint64_t g = (uint64_t)(uintptr_t)(src + off);
            asm volatile("global_load_async_to_lds_b128 %0, %1, off"
                         :: "v"(lds), "v"(g) : "memory");
        }
    };

    issue_jblock(0, 0);
    asm volatile("s_wait_asynccnt 0x0" ::: "memory");
    __syncthreads();

    for (int jb = 0; jb < NJB; ++jb) {
        const int cur = jb & 1;
        if (jb + 1 < NJB) issue_jblock(cur ^ 1, jb + 1);   // prefetch next block

        // 16 x 128 score block: 8 independent WMMA accumulator chains
        v8f acc[JT_];
        #pragma unroll
        for (int t = 0; t < JT_; ++t) acc[t] = (v8f){0.f,0.f,0.f,0.f,0.f,0.f,0.f,0.f};

        const float* jl = &jbuf[cur][0];
        for (int kk = 0; kk < 16; ++kk) {
            #pragma unroll
            for (int t = 0; t < JT_; ++t) {
                v2f bm = *(const v2f*)(jl + (t * 16 + l16) * D_ + kk * 4 + half * 2);
                acc[t] = __builtin_amdgcn_wmma_f32_16x16x4_f32(
                    false, afrag[kk], false, bm, (short)0, acc[t], false, false);
            }
        }

        // LeakyReLU + per-lane online (max, sum-exp) update — no shuffles here.
        #pragma unroll
        for (int t = 0; t < JT_; ++t) {
            #pragma unroll
            for (int r = 0; r < 8; ++r) {
                float x = acc[t][r];
                x = x > 0.0f ? x : ALPHA * x;
                const float mn = fmaxf(m[r], x);
                s[r] = s[r] * __expf(m[r] - mn) + __expf(x - mn);
                m[r] = mn;
            }
        }

        asm volatile("s_wait_asynccnt 0x0" ::: "memory");  // next block landed
        __syncthreads();                                   // all waves done with cur
    }

    // Single cross-lane combine per row: Z = sum_lane s_loc * exp(m_loc - m_glob).
    // Row's 16 j-column slices live across the 16 lanes of this half-wave.
    float z[8];
    #pragma unroll
    for (int r = 0; r < 8; ++r) {
        float mg = m[r];
        mg = fmaxf(mg, __shfl_xor(mg, 8, 16));
        mg = fmaxf(mg, __shfl_xor(mg, 4, 16));
        mg = fmaxf(mg, __shfl_xor(mg, 2, 16));
        mg = fmaxf(mg, __shfl_xor(mg, 1, 16));
        float sc = s[r] * __expf(m[r] - mg);
        sc += __shfl_xor(sc, 8, 16);
        sc += __shfl_xor(sc, 4, 16);
        sc += __shfl_xor(sc, 2, 16);
        sc += __shfl_xor(sc, 1, 16);
        z[r] = sc;
    }

    // att row-sum = (sum_j exp(x_j - m)) / Z with Z the same sum -> z/z (~1.0)
    if (lane == 0)
        #pragma unroll
        for (int r = 0; r < 8; ++r) fsh[wave][r] = z[r] / z[r];
    if (lane == 16)
        #pragma unroll
        for (int r = 0; r < 8; ++r) fsh[wave][8 + r] = z[r] / z[r];
    __syncthreads();

    // out[b, i, hh*D + d] = hp[b,hh,i,d] * factor[i]; 16x64 tile as float4 (b128)
    float* outB = out + ((size_t)b * N_ + i0) * HD_ + hh * D_;
    #pragma unroll
    for (int t = 0; t < 8; ++t) {
        const int idx4 = t * 32 + lane;        // 256 float4 chunks in the tile
        const int row  = idx4 >> 4;
        const int d4   = (idx4 & 15) * 4;
        const float f  = fsh[wave][row];
        float4 v = *(const float4*)(hpBH + (size_t)(i0 + row) * D_ + d4);
        v.x *= f; v.y *= f; v.z *= f; v.w *= f;
        *(float4*)(outB + (size_t)row * HD_ + d4) = v;
    }
}

extern "C" void kernel_launch(void* const* d_in, const int* in_sizes, int n_in,
                              void* d_out, int out_size, void* d_ws, size_t ws_size,
                              hipStream_t stream) {
    const float* h = (const float*)d_in[0];   // [B, N, F] fp32
    const float* W = (const float*)d_in[1];   // [H, F, D] fp32
    float* out = (float*)d_out;               // [B, N, H*D] fp32
    float* hp  = (float*)d_ws;                // [B, H, N, D] fp32 scratch (16.8 MB)

    dim3 grid1(N_ / 16, H_, B_);              // 128 x 4 x 8 waves
    gat_proj_wmma<<<grid1, 32, 0, stream>>>(h, W, hp);

    dim3 grid2(N_ / (16 * 4), H_, B_);        // 32 x 4 x 8 blocks of 4 waves
    gat_attn_rowsum_wmma<<<grid2, 128, 0, stream>>>(hp, out);
}